// QLSTM_65481071398619
// MI455X (gfx1250) — compile-verified
//
#include <hip/hip_runtime.h>
#include <math.h>

typedef float v2f __attribute__((ext_vector_type(2)));
typedef float v8f __attribute__((ext_vector_type(8)));

#define T_STEPS 32
#define BATCH   64
#define DIN     512
#define HID     512
#define PDIM    64
#define NQ      8
#define NBLK    8
#define NTHR    512

__device__ __forceinline__ float2 cmulf(float2 a, float2 b){
  return make_float2(fmaf(a.x, b.x, -a.y * b.y), fmaf(a.x, b.y, a.y * b.x));
}
__device__ __forceinline__ float2 caddf(float2 a, float2 b){
  return make_float2(a.x + b.x, a.y + b.y);
}
__device__ __forceinline__ float shx(float v, int m){ return __shfl_xor(v, m, 32); }
__device__ __forceinline__ float2 shx2(float2 v, int m){
  return make_float2(__shfl_xor(v.x, m, 32), __shfl_xor(v.y, m, 32));
}
__device__ __forceinline__ float wredsum(float v){
  v += shx(v, 16); v += shx(v, 8); v += shx(v, 4); v += shx(v, 2); v += shx(v, 1);
  return v;
}

// Sense-reversing grid barrier across NBLK resident blocks (device scope).
__device__ __forceinline__ void gbar(unsigned* cnt, unsigned* sense){
  __threadfence();
  __syncthreads();
  if (threadIdx.x == 0){
    unsigned gen  = atomicAdd(sense, 0u);
    unsigned prev = atomicAdd(cnt, 1u);
    if (prev == (unsigned)(NBLK - 1)){
      atomicExch(cnt, 0u);
      atomicAdd(sense, 1u);
    } else {
      while (atomicAdd(sense, 0u) == gen) { __builtin_amdgcn_s_sleep(1); }
    }
  }
  __syncthreads();
  __threadfence();
}

// Zero initial hidden state + barrier counters.
__global__ void qlstm_init(float* __restrict__ h, unsigned* __restrict__ cnt){
  int g = blockIdx.x * blockDim.x + threadIdx.x;
  if (g < BATCH * HID) h[g] = 0.f;
  if (g == 0){ cnt[0] = 0u; cnt[1] = 0u; }
}

// Parallel pre-pass: Xpre[t*B+b, p] = x[t,b,:] @ Wx^T + b_proj   (2048x512x64, fp32 WMMA)
__global__ __launch_bounds__(128) void qlstm_xpre(
    const float* __restrict__ x, const float* __restrict__ Wp,
    const float* __restrict__ bp, float* __restrict__ Xpre)
{
  const int wave = (blockIdx.x * 128 + threadIdx.x) >> 5;   // 0..511
  const int lane = threadIdx.x & 31;
  const int tm = wave >> 2, tn = wave & 3;
  const int ln = lane & 15, hi = lane >> 4;
  const float* arow = x  + (size_t)(tm * 16 + ln) * DIN + 2 * hi;
  const float* brow = Wp + (size_t)(tn * 16 + ln) * (DIN + HID) + 2 * hi;
  v8f c;
  const float bn = bp[tn * 16 + ln];
  #pragma unroll
  for (int r = 0; r < 8; ++r) c[r] = bn;
  for (int k = 0; k < DIN; k += 4){
    v2f a = *(const v2f*)(arow + k);
    v2f b = *(const v2f*)(brow + k);
    c = __builtin_amdgcn_wmma_f32_16x16x4_f32(false, a, false, b, (short)0, c, false, false);
  }
  #pragma unroll
  for (int r = 0; r < 8; ++r)
    Xpre[(size_t)(tm * 16 + hi * 8 + r) * PDIM + tn * 16 + ln] = c[r];
}

// Persistent recurrent kernel: NBLK blocks, grid-barrier per phase.
__global__ __launch_bounds__(NTHR, 1) void qlstm_main(
    const float* __restrict__ Wp,   const float* __restrict__ Wtoq,
    const float* __restrict__ Wq2h, const float* __restrict__ fpar,
    const float* __restrict__ ipar, const float* __restrict__ gpar,
    const float* __restrict__ opar, const float* __restrict__ Xpre,
    float* __restrict__ h, float* __restrict__ pre, float* __restrict__ z,
    unsigned* cnt, unsigned* sense, float* __restrict__ out)
{
  const int tid  = threadIdx.x;
  const int lane = tid & 31;
  const int wv   = tid >> 5;            // wave within WG: 0..15
  const int wg   = blockIdx.x;          // 0..7
  const int gw   = wg * 16 + wv;        // global wave: 0..127
  const int gtid = wg * NTHR + tid;     // 0..4095

  // ---- precompute the 128 single-qubit unitaries U = rx@rz@ry into LDS ----
  __shared__ float2 Us[128][4];
  if (tid < 128){
    const int r  = tid >> 5;
    const int gg = (tid >> 4) & 1;
    const int dd = (tid >> 3) & 1;
    const int w  = tid & 7;
    const float* Pb = (r == 0) ? fpar : (r == 1) ? ipar : (r == 2) ? gpar : opar;
    const float* p3 = Pb + (size_t)(((gg * 2) + dd) * NQ + w) * 3;
    const float a = 0.5f * p3[0], b = 0.5f * p3[1], c = 0.5f * p3[2];
    const float ca = cosf(a), sa = sinf(a);
    const float cb = cosf(b), sb = sinf(b);
    const float cc = cosf(c), sc = sinf(c);
    // M = rz @ ry
    const float2 M00 = make_float2( ca * cb, -ca * sb);
    const float2 M01 = make_float2(-sa * cb,  sa * sb);
    const float2 M10 = make_float2( sa * cb,  sa * sb);
    const float2 M11 = make_float2( ca * cb,  ca * sb);
    // U = rx @ M, rx = [[cc, -i*sc],[-i*sc, cc]];  (-i*sc)*(x,y) = (sc*y, -sc*x)
    Us[tid][0] = make_float2(cc * M00.x + sc * M10.y, cc * M00.y - sc * M10.x);
    Us[tid][1] = make_float2(cc * M01.x + sc * M11.y, cc * M01.y - sc * M11.x);
    Us[tid][2] = make_float2(sc * M00.y + cc * M10.x, cc * M10.y - sc * M00.x);
    Us[tid][3] = make_float2(sc * M01.y + cc * M11.x, cc * M11.y - sc * M01.x);
  }
  __syncthreads();

  float creg[8];
  #pragma unroll
  for (int k = 0; k < 8; ++k) creg[k] = 0.f;

  for (int t = 0; t < T_STEPS; ++t){
    // ---------- Phase A: pre = Xpre[t] + h @ Wh^T  (16 fp32-WMMA tiles) ----------
    if (wv < 2){
      const int ti = wg * 2 + wv;          // tile 0..15
      const int tm = ti >> 2, tn = ti & 3;
      const int ln = lane & 15, hb = lane >> 4;
      const float* arow = h  + (size_t)(tm * 16 + ln) * HID + 2 * hb;
      const float* brow = Wp + (size_t)(tn * 16 + ln) * (DIN + HID) + DIN + 2 * hb;
      v8f c;
      #pragma unroll
      for (int r = 0; r < 8; ++r)
        c[r] = Xpre[(size_t)(t * BATCH + tm * 16 + hb * 8 + r) * PDIM + tn * 16 + ln];
      for (int k = 0; k < HID; k += 4){
        v2f a = *(const v2f*)(arow + k);
        v2f b = *(const v2f*)(brow + k);
        c = __builtin_amdgcn_wmma_f32_16x16x4_f32(false, a, false, b, (short)0, c, false, false);
      }
      #pragma unroll
      for (int r = 0; r < 8; ++r)
        pre[(size_t)(tm * 16 + hb * 8 + r) * PDIM + tn * 16 + ln] = c[r];
    }
    gbar(cnt, sense);

    // ---------- Phase B: qin + 2 statevector sims per wave ----------
    {
      const int b = gw & 63;
      const float* prow = pre + (size_t)b * PDIM;
      float qin[NQ];
      #pragma unroll
      for (int q = 0; q < NQ; ++q){
        float v = fmaf(prow[lane], Wtoq[q * PDIM + lane],
                       prow[lane + 32] * Wtoq[q * PDIM + lane + 32]);
        qin[q] = wredsum(v);
      }
      #pragma unroll 1
      for (int si = 0; si < 2; ++si){
        const int s = gw + si * 128;      // sim id 0..255
        const int r = s >> 6;             // branch 0..3
        float xc[NQ];
        #pragma unroll
        for (int q = 0; q < NQ; ++q) xc[q] = qin[q];

        #pragma unroll 1
        for (int gg = 0; gg < 2; ++gg){
          float chv[NQ], shv[NQ];
          #pragma unroll
          for (int q = 0; q < NQ; ++q){
            chv[q] = cosf(0.5f * xc[q]);
            shv[q] = sinf(0.5f * xc[q]);
          }
          // state preparation: st[k] = (-i)^popc(k) * prod_q (bit? sin : cos)
          float2 st[8];
          #pragma unroll
          for (int j = 0; j < 8; ++j){
            const int k = lane * 8 + j;
            float m = 1.f;
            #pragma unroll
            for (int q = 0; q < NQ; ++q) m *= ((k >> (7 - q)) & 1) ? shv[q] : chv[q];
            const int pp = __popc(k) & 3;
            st[j] = make_float2((pp == 0) ? m : ((pp == 2) ? -m : 0.f),
                                (pp == 1) ? -m : ((pp == 3) ? m : 0.f));
          }
          #pragma unroll 1
          for (int dd = 0; dd < 2; ++dd){
            const int mbase = ((r * 2 + gg) * 2 + dd) * 8;
            // single-qubit gates, wire w acts on amplitude bit (7-w)
            #pragma unroll
            for (int w = 0; w < 8; ++w){
              const float2 u00 = Us[mbase + w][0], u01 = Us[mbase + w][1];
              const float2 u10 = Us[mbase + w][2], u11 = Us[mbase + w][3];
              const int bq = 7 - w;
              if (bq >= 3){               // lane bit: pair via shfl_xor
                const int lm = 1 << (bq - 3);
                const int mybit = (lane >> (bq - 3)) & 1;
                #pragma unroll
                for (int j = 0; j < 8; ++j){
                  float2 pv = shx2(st[j], lm);
                  float2 lo = caddf(cmulf(u00, st[j]), cmulf(u01, pv));
                  float2 hi = caddf(cmulf(u10, pv), cmulf(u11, st[j]));
                  st[j] = mybit ? hi : lo;
                }
              } else {                    // in-register pair
                const int m2 = 1 << bq;
                #pragma unroll
                for (int j0 = 0; j0 < 8; ++j0){
                  if ((j0 & m2) == 0){
                    float2 a0 = st[j0], a1 = st[j0 | m2];
                    st[j0]      = caddf(cmulf(u00, a0), cmulf(u01, a1));
                    st[j0 | m2] = caddf(cmulf(u10, a0), cmulf(u11, a1));
                  }
                }
              }
            }
            // CNOT ladder: control bit (7-c), target bit (6-c)
            #pragma unroll
            for (int c2 = 0; c2 < 7; ++c2){
              const int bc = 7 - c2, bt = 6 - c2;
              if (bt >= 3){               // both lane bits
                const int lm = 1 << (bt - 3);
                const int cb = (lane >> (bc - 3)) & 1;
                #pragma unroll
                for (int j = 0; j < 8; ++j){
                  float2 sw = shx2(st[j], lm);
                  st[j] = cb ? sw : st[j];
                }
              } else if (bc == 3){        // control = lane bit 0, target = in-lane bit 2
                const int cb = lane & 1;
                #pragma unroll
                for (int j = 0; j < 4; ++j){
                  float2 a = st[j], bv = st[j + 4];
                  st[j]     = cb ? bv : a;
                  st[j + 4] = cb ? a  : bv;
                }
              } else {                    // both in-lane bits
                const int mT = 1 << bt;
                #pragma unroll
                for (int j = 0; j < 8; ++j){
                  if (((j >> bc) & 1) && !((j >> bt) & 1)){
                    float2 tmp = st[j]; st[j] = st[j | mT]; st[j | mT] = tmp;
                  }
                }
              }
            }
          }
          // measure <Z_q> = sum_k |st[k]|^2 * (1 - 2*bit_{7-q}(k))
          float acc[NQ];
          #pragma unroll
          for (int q = 0; q < NQ; ++q) acc[q] = 0.f;
          #pragma unroll
          for (int j = 0; j < 8; ++j){
            const int k = lane * 8 + j;
            const float p = fmaf(st[j].x, st[j].x, st[j].y * st[j].y);
            #pragma unroll
            for (int q = 0; q < NQ; ++q) acc[q] += ((k >> (7 - q)) & 1) ? -p : p;
          }
          #pragma unroll
          for (int q = 0; q < NQ; ++q) xc[q] = wredsum(acc[q]);
        }
        if (lane == 0){
          #pragma unroll
          for (int q = 0; q < NQ; ++q) z[s * NQ + q] = xc[q];
        }
      }
    }
    gbar(cnt, sense);

    // ---------- Phase C: gate mix (K=8) + LSTM cell update ----------
    #pragma unroll
    for (int k = 0; k < 8; ++k){
      const int e  = gtid + k * (NBLK * NTHR);
      const int bb = e >> 9;
      const int hh = e & 511;
      const float* wrow = Wq2h + (size_t)hh * NQ;
      float az[4];
      #pragma unroll
      for (int r = 0; r < 4; ++r){
        const float* zr = z + (size_t)(r * 64 + bb) * NQ;
        float s2 = 0.f;
        #pragma unroll
        for (int q = 0; q < NQ; ++q) s2 = fmaf(zr[q], wrow[q], s2);
        az[r] = s2;
      }
      const float fv = 1.f / (1.f + expf(-az[0]));
      const float iv = 1.f / (1.f + expf(-az[1]));
      const float gv = tanhf(az[2]);
      const float ov = 1.f / (1.f + expf(-az[3]));
      const float cn = fmaf(fv, creg[k], iv * gv);
      creg[k] = cn;
      const float hv = ov * tanhf(cn);
      h[e] = hv;
      out[(size_t)t * (BATCH * HID) + e] = hv;
      if (t == T_STEPS - 1){
        out[(size_t)T_STEPS * BATCH * HID + e] = hv;                 // hx
        out[(size_t)T_STEPS * BATCH * HID + BATCH * HID + e] = cn;   // cx
      }
    }
    gbar(cnt, sense);
  }
}

extern "C" void kernel_launch(void* const* d_in, const int* in_sizes, int n_in,
                              void* d_out, int out_size, void* d_ws, size_t ws_size,
                              hipStream_t stream) {
  (void)in_sizes; (void)n_in; (void)out_size; (void)ws_size;
  const float* x    = (const float*)d_in[0];   // (32,64,512)
  const float* Wp   = (const float*)d_in[1];   // (64,1024)
  const float* bp   = (const float*)d_in[2];   // (64,)
  const float* Wtoq = (const float*)d_in[3];   // (8,64)
  const float* Wq2h = (const float*)d_in[4];   // (512,8)
  const float* fpar = (const float*)d_in[5];   // (2,2,8,3)
  const float* ipar = (const float*)d_in[6];
  const float* gpar = (const float*)d_in[7];
  const float* opar = (const float*)d_in[8];
  float* out = (float*)d_out;

  float*    wsf   = (float*)d_ws;
  unsigned* cnt   = (unsigned*)d_ws;           // cnt[0]=count, cnt[1]=sense
  float*    xpre  = wsf + 16;                                   // 2048*64
  float*    hbuf  = xpre + (size_t)T_STEPS * BATCH * PDIM;      // 64*512
  float*    prebf = hbuf + (size_t)BATCH * HID;                 // 64*64
  float*    zbuf  = prebf + (size_t)BATCH * PDIM;               // 256*8

  qlstm_init<<<(BATCH * HID + 255) / 256, 256, 0, stream>>>(hbuf, cnt);
  qlstm_xpre<<<128, 128, 0, stream>>>(x, Wp, bp, xpre);
  qlstm_main<<<NBLK, NTHR, 0, stream>>>(Wp, Wtoq, Wq2h, fpar, ipar, gpar, opar,
                                        xpre, hbuf, prebf, zbuf,
                                        cnt, cnt + 1, out);
}